// FourParameterExponentialFamilyParticleFilter_42666205118529
// MI455X (gfx1250) — compile-verified
//
#include <hip/hip_runtime.h>
#include <hip/hip_bf16.h>
#include <math.h>

// Problem constants (match reference)
#define BB     64
#define TT     1024
#define DD     256
#define KK     128
#define ALPHA  1.0f
#define EPSV   1e-3f
#define PI_F   3.14159265358979323846f

typedef __attribute__((ext_vector_type(8))) float v8f;
typedef __attribute__((ext_vector_type(2))) float v2f;

// ---------------------------------------------------------------------------
// Async copy: one 4-byte element per lane, global -> LDS, tracked by ASYNCcnt.
// (CDNA5 ISA: GLOBAL_LOAD_ASYNC_TO_LDS_B32, GV mode)
// ---------------------------------------------------------------------------
__device__ __forceinline__ void async_copy_f32(unsigned ldsOff, const float* gptr)
{
    asm volatile("global_load_async_to_lds_b32 %0, %1, off"
                 :: "v"(ldsOff), "v"((unsigned long long)(uintptr_t)gptr)
                 : "memory");
}

// ---------------------------------------------------------------------------
// Wave-level f32 reduction via V_WMMA_F32_16X16X4_F32.
// A 16x4 layout (ISA 7.12.2): lanes 0-15 hold {K=0,K=1}, lanes 16-31 {K=2,K=3}.
// With a={v,0} and B=ones: D[m][n] = v_m + v_{m+16}. Summing a lane's 8 C
// VGPRs gives a half sum (M=0..7 on lanes<16, M=8..15 on lanes>=16); one
// xor-16 shuffle completes the 32-lane total (returned in every lane).
// ---------------------------------------------------------------------------
__device__ __forceinline__ float wave_reduce_wmma_f32(float v)
{
    v2f a;  a[0] = v;     a[1] = 0.0f;
    v2f bm; bm[0] = 1.0f; bm[1] = 1.0f;
    v8f c = {};
    c = __builtin_amdgcn_wmma_f32_16x16x4_f32(
        /*neg_a=*/false, a, /*neg_b=*/false, bm,
        /*c_mod=*/(short)0, c, /*reuse_a=*/false, /*reuse_b=*/false);
    float s = c[0] + c[1] + c[2] + c[3] + c[4] + c[5] + c[6] + c[7];
    s += __shfl_xor(s, 16, 32);
    return s;
}

// ---------------------------------------------------------------------------
// Kernel A: per-batch rank computation + counting sort of timesteps by cluster
// grid = B, block = 128
// ---------------------------------------------------------------------------
__global__ void crp_rank_sort_kernel(const int* __restrict__ z,
                                     int* __restrict__ rank,     // [B*T]
                                     int* __restrict__ ord,      // [B*T]
                                     int* __restrict__ offs)     // [B*(K+1)]
{
    const int b   = blockIdx.x;
    const int tid = threadIdx.x;
    __shared__ int counts[KK];
    __shared__ int ex[KK + 1];

    if (tid < KK) counts[tid] = 0;
    __syncthreads();

    if (tid == 0) {
        const int* zb = z + b * TT;
        int* rb = rank + b * TT;
        for (int t = 0; t < TT; ++t) {
            int k = zb[t];
            int r = counts[k];
            counts[k] = r + 1;
            rb[t] = r;
        }
        int acc = 0;
        for (int k = 0; k < KK; ++k) { ex[k] = acc; acc += counts[k]; }
        ex[KK] = acc;
    }
    __syncthreads();

    if (tid <= KK) offs[b * (KK + 1) + tid] = ex[tid];

    const int* zb = z + b * TT;
    const int* rb = rank + b * TT;
    int* ob = ord + b * TT;
    for (int t = tid; t < TT; t += blockDim.x) {
        int k = zb[t];
        ob[ex[k] + rb[t]] = t;
    }
}

// ---------------------------------------------------------------------------
// Kernel B: main compute. One block per (b,k); one thread per dim d.
// Sufficient statistics in registers; X rows staged through LDS with
// double-buffered async-to-LDS loads. The lgamma pair is maintained by the
// Gamma recurrence (lgamma(an+1) = lgamma(an) + log(an)) so the hot loop
// pays one logf instead of two ocml lgamma calls per step.
// grid = B*K, block = 256
// ---------------------------------------------------------------------------
__global__ void crp_hurdle_main_kernel(const float* __restrict__ X,
                                       const float* __restrict__ loc,
                                       const float* __restrict__ log_mean_conc,
                                       const float* __restrict__ log_conc,
                                       const float* __restrict__ log_scale,
                                       const float* __restrict__ sparse_prior_logit,
                                       const int* __restrict__ ord,
                                       const int* __restrict__ offs,
                                       float* __restrict__ partial) // [B*K]
{
    const int b = blockIdx.x / KK;
    const int k = blockIdx.x % KK;
    const int d = threadIdx.x;

    const int base = offs[b * (KK + 1) + k];
    const int n    = offs[b * (KK + 1) + k + 1] - base;

    // per-dim hyperparameters
    const float locv   = loc[d];
    const float kappa0 = __expf(fminf(log_mean_conc[d], 12.0f));
    const float a_nig  = __expf(log_conc[d]);
    const float b_nig  = __expf(log_scale[d]);
    const float nu     = 2.0f * a_nig + 3.0f;
    const float a0     = nu + 1.0f;
    const float b0     = (nu + 1.0f) * __expf(sparse_prior_logit[d]);

    float m1 = 0.0f, ss1 = 0.0f, ss2 = 0.0f;
    float lsum = 0.0f;

    const float* Xb   = X + (size_t)b * TT * DD;
    const int*   ordb = ord + b * TT + base;

    __shared__ float rowbuf[2][DD];

    if (n > 0) {
        const unsigned lds0 = (unsigned)(uintptr_t)&rowbuf[0][d];
        const unsigned lds1 = (unsigned)(uintptr_t)&rowbuf[1][d];

        // lgamma pair seeded once; advanced by recurrence in the loop
        float an   = a_nig;                       // = a_nig + 0.5*m1
        float lgA  = lgammaf(an);                 // lgamma(an)
        float lgA5 = lgammaf(an + 0.5f);          // lgamma(an + 0.5)

        // prime the pipeline: stage row 0
        {
            int t0 = ordb[0];
            async_copy_f32(lds0, Xb + (size_t)t0 * DD + d);
        }

        for (int j = 0; j < n; ++j) {
            if (j + 1 < n) {
                int tn = ordb[j + 1];
                async_copy_f32(((j + 1) & 1) ? lds1 : lds0,
                               Xb + (size_t)tn * DD + d);
                // in-order async completion: <=1 outstanding => row j landed
                asm volatile("s_wait_asynccnt 0x1" ::: "memory");
            } else {
                asm volatile("s_wait_asynccnt 0x0" ::: "memory");
            }
            const int   t   = ordb[j];
            const float x   = rowbuf[j & 1][d];
            const float nzf = (x > EPSV) ? 1.0f : 0.0f;
            const float yv  = logf(fmaxf(x, EPSV));
            const float ckf = (float)j;

            const float ldenom  = logf(a0 + b0 + ckf);
            const float lp_zero = logf(b0 + (ckf - m1)) - ldenom;

            // NIG posterior -> Student-t predictive on log x
            const float kn  = kappa0 + m1;
            const float mn  = (kappa0 * locv + ss1) / kn;
            const float bn  = b_nig + 0.5f * (ss2 + kappa0 * locv * locv - kn * mn * mn);
            const float df  = 2.0f * an;
            const float sc2 = bn * (kn + 1.0f) / (an * kn);
            const float dy  = yv - mn;
            const float tl  = (lgA5 - lgA)
                            - 0.5f * logf(df * PI_F * sc2)
                            - 0.5f * (df + 1.0f) * log1pf(dy * dy / (df * sc2));
            const float lp_nz = logf(a0 + m1) - ldenom + tl - yv;

            lsum += (nzf > 0.0f) ? lp_nz : lp_zero;

            // CRP prior term (lane 0 carries it)
            if (d == 0) {
                float num = (ckf > 0.0f) ? logf(fmaxf(ckf, 1.0f)) : logf(ALPHA);
                lsum += num - logf((float)t + ALPHA);
            }

            // conjugate updates (registers only) + lgamma recurrence:
            // on nz: an += 0.5; lgamma(an') = lgA5; lgamma(an'+0.5) = lgA + log(an)
            const float lgNext = lgA + logf(an);
            if (nzf > 0.0f) {
                lgA  = lgA5;
                lgA5 = lgNext;
                an  += 0.5f;
            }
            m1  += nzf;
            ss1 += nzf * yv;
            ss2 += nzf * yv * yv;
        }
    }

    // block reduction over D: 256 -> 32 via LDS, 32 -> 1 via f32 WMMA
    __shared__ float red[DD];
    red[d] = lsum;
    __syncthreads();
    if (d < 32) {                       // wave-uniform: wave 0, EXEC all ones
        float v = 0.0f;
#pragma unroll
        for (int i = 0; i < DD / 32; ++i) v += red[d + 32 * i];
        float tot = wave_reduce_wmma_f32(v);
        if (d == 0) partial[b * KK + k] = tot;
    }
}

// ---------------------------------------------------------------------------
// Kernel C: per-batch reduction of K partials -> out[b] = -lp.
// Single wave; 128 -> 32 via coalesced loads, 32 -> 1 via f32 WMMA.
// grid = B, block = 32
// ---------------------------------------------------------------------------
__global__ void crp_reduce_kernel(const float* __restrict__ partial,
                                  float* __restrict__ out)
{
    const int b    = blockIdx.x;
    const int lane = threadIdx.x;

    float v = 0.0f;
#pragma unroll
    for (int i = 0; i < KK / 32; ++i) v += partial[b * KK + lane + 32 * i];

    float tot = wave_reduce_wmma_f32(v);
    if (lane == 0) out[b] = -tot;
}

// ---------------------------------------------------------------------------
// Host-side launcher
// ---------------------------------------------------------------------------
extern "C" void kernel_launch(void* const* d_in, const int* in_sizes, int n_in,
                              void* d_out, int out_size, void* d_ws, size_t ws_size,
                              hipStream_t stream)
{
    const float* X   = (const float*)d_in[0];
    const float* loc = (const float*)d_in[1];
    const float* lmc = (const float*)d_in[2];
    const float* lc  = (const float*)d_in[3];
    const float* ls  = (const float*)d_in[4];
    const float* spl = (const float*)d_in[5];
    const int*   z   = (const int*)d_in[6];
    float* out = (float*)d_out;

    char* ws = (char*)d_ws;
    int*   rank    = (int*)(ws);                                   // B*T ints
    int*   ord     = (int*)(ws + (size_t)BB * TT * 4);             // B*T ints
    int*   offs    = (int*)(ws + (size_t)2 * BB * TT * 4);         // B*(K+1) ints
    float* partial = (float*)(ws + (size_t)2 * BB * TT * 4
                                 + (size_t)BB * (KK + 1) * 4);     // B*K floats

    crp_rank_sort_kernel<<<BB, 128, 0, stream>>>(z, rank, ord, offs);
    crp_hurdle_main_kernel<<<BB * KK, DD, 0, stream>>>(X, loc, lmc, lc, ls, spl,
                                                       ord, offs, partial);
    crp_reduce_kernel<<<BB, 32, 0, stream>>>(partial, out);
}